// ICP_1400159339134
// MI455X (gfx1250) — compile-verified
//
#include <hip/hip_runtime.h>

// ============================================================================
// ICP on MI455X (gfx1250): fused WMMA nearest-neighbor + deterministic fit.
//
// Hot loop: V_WMMA_F32_16X16X4_F32 scores a 16(src) x 16(dst) tile per
// instruction using the K=4 trick  score = [-2X,-2Y,-2Z,|d|^2] . [x,y,z,1].
// argmin_j(score) == argmin_j(|s-d_j|^2); |s|^2 is added back only for err.
// ============================================================================

typedef float v2f __attribute__((ext_vector_type(2)));
typedef float v8f __attribute__((ext_vector_type(8)));

static constexpr int NP     = 8192;
static constexpr int NTILE  = NP / 16;      // 512 src tiles
static constexpr int NCHUNK = NP / 16;      // 512 dst chunks
static constexpr int WPB    = 8;            // waves per block (wave32)
static constexpr int BLK_A  = NTILE / WPB;  // 64 blocks for the NN kernel
static constexpr int ITERS  = 20;
static constexpr float TOL  = 1e-3f;

// Workspace layout (float offsets). Total ~235 KB.
static constexpr int OFF_SX    = 0;                      // src x [8192]
static constexpr int OFF_SY    = OFF_SX + NP;            // src y
static constexpr int OFF_SZ    = OFF_SY + NP;            // src z
static constexpr int OFF_BP    = OFF_SZ + NP;            // packed B operand: v2f[NCHUNK*32]
static constexpr int OFF_PART  = OFF_BP + NCHUNK * 32 * 2; // block partials [BLK_A*16]
static constexpr int OFF_STATE = OFF_PART + BLK_A * 16;  // [0]=prev_err [1]=done

// ----------------------------------------------------------------------------
// Horn quaternion solve == Kabsch SVD with reflection fix (optimal proper R).
// H[a*3+b] = sum_i (s_i - cS)[a] * (b_i - cB)[b]
// ----------------------------------------------------------------------------
__device__ inline void horn_fit(const float H[9], const float cA[3], const float cB[3],
                                float R[9], float t[3]) {
  const float Sxx = H[0], Sxy = H[1], Sxz = H[2];
  const float Syx = H[3], Syy = H[4], Syz = H[5];
  const float Szx = H[6], Szy = H[7], Szz = H[8];
  float Nq[16] = {
    Sxx + Syy + Szz, Syz - Szy,        Szx - Sxz,        Sxy - Syx,
    Syz - Szy,       Sxx - Syy - Szz,  Sxy + Syx,        Szx + Sxz,
    Szx - Sxz,       Sxy + Syx,       -Sxx + Syy - Szz,  Syz + Szy,
    Sxy - Syx,       Szx + Sxz,        Syz + Szy,       -Sxx - Syy + Szz };
  // Shift so the largest eigenvalue is dominant in magnitude (Gershgorin bound).
  float shift = 1.0f;
  #pragma unroll
  for (int i = 0; i < 16; ++i) shift += fabsf(Nq[i]);
  Nq[0] += shift; Nq[5] += shift; Nq[10] += shift; Nq[15] += shift;
  float q[4] = {1.f, 0.f, 0.f, 0.f};
  for (int it = 0; it < 64; ++it) {            // fixed-count -> deterministic
    float w[4];
    #pragma unroll
    for (int r = 0; r < 4; ++r)
      w[r] = Nq[r*4+0]*q[0] + Nq[r*4+1]*q[1] + Nq[r*4+2]*q[2] + Nq[r*4+3]*q[3];
    float inv = rsqrtf(w[0]*w[0] + w[1]*w[1] + w[2]*w[2] + w[3]*w[3] + 1e-30f);
    #pragma unroll
    for (int r = 0; r < 4; ++r) q[r] = w[r] * inv;
  }
  const float qw = q[0], qx = q[1], qy = q[2], qz = q[3];
  R[0] = 1.f - 2.f*(qy*qy + qz*qz); R[1] = 2.f*(qx*qy - qw*qz); R[2] = 2.f*(qx*qz + qw*qy);
  R[3] = 2.f*(qx*qy + qw*qz); R[4] = 1.f - 2.f*(qx*qx + qz*qz); R[5] = 2.f*(qy*qz - qw*qx);
  R[6] = 2.f*(qx*qz - qw*qy); R[7] = 2.f*(qy*qz + qw*qx); R[8] = 1.f - 2.f*(qx*qx + qy*qy);
  #pragma unroll
  for (int r = 0; r < 3; ++r)
    t[r] = cB[r] - (R[r*3+0]*cA[0] + R[r*3+1]*cA[1] + R[r*3+2]*cA[2]);
}

// ----------------------------------------------------------------------------
// Init: src SoA from A; pack dst into WMMA-B operand order.
// B operand layout assumed (mirrors ISA A/C tables): VGPR0 lanes0-15 -> K0,
// lanes16-31 -> K2; VGPR1 lanes0-15 -> K1, lanes16-31 -> K3; N = lane%16.
// Rows: K0=-2X, K1=-2Y, K2=-2Z, K3=|d|^2.
// ----------------------------------------------------------------------------
__global__ __launch_bounds__(256) void icp_init(const float* __restrict__ A,
                                                const float* __restrict__ B,
                                                float* __restrict__ ws) {
  int p = blockIdx.x * blockDim.x + threadIdx.x;
  if (p < NP) {
    ws[OFF_SX + p] = A[3*p + 0];
    ws[OFF_SY + p] = A[3*p + 1];
    ws[OFF_SZ + p] = A[3*p + 2];
    float X = B[3*p + 0], Y = B[3*p + 1], Z = B[3*p + 2];
    float q = X*X + Y*Y + Z*Z;
    int c = p >> 4, n = p & 15;
    v2f* Bp = (v2f*)(ws + OFF_BP);
    v2f lo; lo.x = -2.f*X; lo.y = -2.f*Y;    // lane n   : (K0, K1)
    v2f hi; hi.x = -2.f*Z; hi.y =  q;        // lane n+16: (K2, K3)
    Bp[c*32 + n]      = lo;
    Bp[c*32 + 16 + n] = hi;
  }
  if (p == 0) { ws[OFF_STATE + 0] = 0.f; ws[OFF_STATE + 1] = 0.f; }
}

// ----------------------------------------------------------------------------
// NN + fused reductions. One wave = one 16-row src tile; streams all 8192 dst
// points in 512 WMMA chunks. Per-block partials (deterministic, no atomics):
// [0]=sum dist, [1..3]=sum B[idx], [4..12]=sum s (x) B[idx], [13..15]=sum s.
// ----------------------------------------------------------------------------
__global__ __launch_bounds__(256) void icp_nn(const float* __restrict__ B,
                                              float* __restrict__ ws) {
  __shared__ float red[WPB * 16];
  const int tid  = threadIdx.x;
  const int wave = tid >> 5;
  const int lane = tid & 31;
  const int tile = blockIdx.x * WPB + wave;
  const int r    = lane & 15;
  const int hi   = lane >> 4;
  const int i    = tile * 16 + r;

  const float sx = ws[OFF_SX + i];
  const float sy = ws[OFF_SY + i];
  const float sz = ws[OFF_SZ + i];
  const float s2 = sx*sx + sy*sy + sz*sz;

  // A operand (16x4 f32): VGPR0 lanes0-15->K0(x), lanes16-31->K2(z);
  //                       VGPR1 lanes0-15->K1(y), lanes16-31->K3(1).
  v2f a;
  a.x = hi ? sz : sx;
  a.y = hi ? 1.0f : sy;

  const v2f* __restrict__ Bp = (const v2f*)(ws + OFF_BP);

  float minv[8];
  int   mini[8];
  #pragma unroll
  for (int v = 0; v < 8; ++v) { minv[v] = 3.0e38f; mini[v] = 0; }

  for (int c = 0; c < NCHUNK; ++c) {
    v2f b = Bp[c*32 + lane];
    v8f z = {};
    // D = A x B (+0): 16x16 scores, D[v] at this lane = score(row M(v,lane), dst c*16 + lane%16)
    v8f d = __builtin_amdgcn_wmma_f32_16x16x4_f32(false, a, false, b,
                                                  (short)0, z, false, false);
    const int jc = (c << 4) + r;
    #pragma unroll
    for (int v = 0; v < 8; ++v) {
      bool lt = d[v] < minv[v];
      minv[v] = lt ? d[v] : minv[v];
      mini[v] = lt ? jc   : mini[v];
    }
  }

  // Cross-lane argmin within each 16-lane half (xor<16 stays inside half).
  #pragma unroll
  for (int m = 8; m >= 1; m >>= 1) {
    #pragma unroll
    for (int v = 0; v < 8; ++v) {
      float ov = __shfl_xor(minv[v], m, 32);
      int   oi = __shfl_xor(mini[v], m, 32);
      bool take = (ov < minv[v]) || (ov == minv[v] && oi < mini[v]);
      minv[v] = take ? ov : minv[v];
      mini[v] = take ? oi : mini[v];
    }
  }

  // Writers: lanes 0..7 -> rows 0..7 (half A), lanes 24..31 -> rows 8..15 (half B).
  const bool writer = (lane < 8) || (lane >= 24);
  const int  vv = r & 7;
  float c16[16];
  #pragma unroll
  for (int k = 0; k < 16; ++k) c16[k] = 0.f;
  if (writer) {
    const float score = minv[vv];
    const int   j     = mini[vv];
    const float dd = sqrtf(fmaxf(score + s2, 0.f) + 1e-12f);
    const float bx = B[3*j + 0], by = B[3*j + 1], bz = B[3*j + 2];
    c16[0]  = dd;
    c16[1]  = bx;    c16[2]  = by;    c16[3]  = bz;
    c16[4]  = sx*bx; c16[5]  = sx*by; c16[6]  = sx*bz;
    c16[7]  = sy*bx; c16[8]  = sy*by; c16[9]  = sy*bz;
    c16[10] = sz*bx; c16[11] = sz*by; c16[12] = sz*bz;
    c16[13] = sx;    c16[14] = sy;    c16[15] = sz;
  }
  // Full-wave tree sum (deterministic), then per-block LDS combine.
  #pragma unroll
  for (int m = 16; m >= 1; m >>= 1) {
    #pragma unroll
    for (int k = 0; k < 16; ++k) c16[k] += __shfl_xor(c16[k], m, 32);
  }
  if (lane == 0) {
    #pragma unroll
    for (int k = 0; k < 16; ++k) red[wave*16 + k] = c16[k];
  }
  __syncthreads();
  if (tid < 16) {
    float s = 0.f;
    for (int w = 0; w < WPB; ++w) s += red[w*16 + tid];
    ws[OFF_PART + blockIdx.x*16 + tid] = s;
  }
}

// ----------------------------------------------------------------------------
// Fit + apply (1 block). Mirrors jax.lax.scan carry semantics:
//   converged = |prev_err - err| < TOL; src/prev_err masked by OLD done;
//   done |= converged.
// ----------------------------------------------------------------------------
__global__ __launch_bounds__(256) void icp_fit(float* __restrict__ ws) {
  __shared__ float sRT[12];
  __shared__ int   sApply;
  if (threadIdx.x == 0) {
    float P[16];
    #pragma unroll
    for (int k = 0; k < 16; ++k) P[k] = 0.f;
    for (int b = 0; b < BLK_A; ++b)
      for (int k = 0; k < 16; ++k) P[k] += ws[OFF_PART + b*16 + k];
    const float Nf = (float)NP;
    const float cS[3] = { P[13]/Nf, P[14]/Nf, P[15]/Nf };
    const float cB[3] = { P[1]/Nf,  P[2]/Nf,  P[3]/Nf  };
    float H[9];
    #pragma unroll
    for (int aa = 0; aa < 3; ++aa)
      #pragma unroll
      for (int bb = 0; bb < 3; ++bb)
        H[aa*3+bb] = P[4 + aa*3 + bb] - Nf * cS[aa] * cB[bb];
    float R[9], t[3];
    horn_fit(H, cS, cB, R, t);
    const float err  = P[0] / Nf;
    const float prev = ws[OFF_STATE + 0];
    const float done = ws[OFF_STATE + 1];
    const int apply = (done == 0.f);
    const int conv  = fabsf(prev - err) < TOL;
    if (apply) ws[OFF_STATE + 0] = err;
    if (conv)  ws[OFF_STATE + 1] = 1.f;
    sApply = apply;
    #pragma unroll
    for (int k = 0; k < 9; ++k) sRT[k] = R[k];
    sRT[9] = t[0]; sRT[10] = t[1]; sRT[11] = t[2];
  }
  __syncthreads();
  if (sApply) {
    const float R0 = sRT[0], R1 = sRT[1], R2 = sRT[2];
    const float R3 = sRT[3], R4 = sRT[4], R5 = sRT[5];
    const float R6 = sRT[6], R7 = sRT[7], R8 = sRT[8];
    const float t0 = sRT[9], t1 = sRT[10], t2 = sRT[11];
    for (int i = threadIdx.x; i < NP; i += 256) {
      const float x = ws[OFF_SX + i], y = ws[OFF_SY + i], z = ws[OFF_SZ + i];
      ws[OFF_SX + i] = R0*x + R1*y + R2*z + t0;
      ws[OFF_SY + i] = R3*x + R4*y + R5*z + t1;
      ws[OFF_SZ + i] = R6*x + R7*y + R8*z + t2;
    }
  }
}

// ----------------------------------------------------------------------------
// Final: T = best_fit_transform(A, src_final), written as 4x4 row-major.
// ----------------------------------------------------------------------------
__global__ __launch_bounds__(256) void icp_final(const float* __restrict__ A,
                                                 const float* __restrict__ ws,
                                                 float* __restrict__ out) {
  __shared__ float red[WPB * 15];
  const int tid  = threadIdx.x;
  const int wave = tid >> 5;
  const int lane = tid & 31;
  float acc[15];
  #pragma unroll
  for (int k = 0; k < 15; ++k) acc[k] = 0.f;
  for (int i = tid; i < NP; i += 256) {
    const float ax = A[3*i+0], ay = A[3*i+1], az = A[3*i+2];
    const float sx = ws[OFF_SX + i], sy = ws[OFF_SY + i], sz = ws[OFF_SZ + i];
    acc[0] += ax; acc[1] += ay; acc[2] += az;
    acc[3] += sx; acc[4] += sy; acc[5] += sz;
    acc[6]  += ax*sx; acc[7]  += ax*sy; acc[8]  += ax*sz;
    acc[9]  += ay*sx; acc[10] += ay*sy; acc[11] += ay*sz;
    acc[12] += az*sx; acc[13] += az*sy; acc[14] += az*sz;
  }
  #pragma unroll
  for (int m = 16; m >= 1; m >>= 1) {
    #pragma unroll
    for (int k = 0; k < 15; ++k) acc[k] += __shfl_xor(acc[k], m, 32);
  }
  if (lane == 0) {
    #pragma unroll
    for (int k = 0; k < 15; ++k) red[wave*15 + k] = acc[k];
  }
  __syncthreads();
  if (tid == 0) {
    float S[15];
    #pragma unroll
    for (int k = 0; k < 15; ++k) S[k] = 0.f;
    for (int w = 0; w < WPB; ++w)
      for (int k = 0; k < 15; ++k) S[k] += red[w*15 + k];
    const float Nf = (float)NP;
    const float cA[3] = { S[0]/Nf, S[1]/Nf, S[2]/Nf };
    const float cS[3] = { S[3]/Nf, S[4]/Nf, S[5]/Nf };
    float H[9];
    #pragma unroll
    for (int aa = 0; aa < 3; ++aa)
      #pragma unroll
      for (int bb = 0; bb < 3; ++bb)
        H[aa*3+bb] = S[6 + aa*3 + bb] - Nf * cA[aa] * cS[bb];
    float R[9], t[3];
    horn_fit(H, cA, cS, R, t);
    out[0]  = R[0]; out[1]  = R[1]; out[2]  = R[2]; out[3]  = t[0];
    out[4]  = R[3]; out[5]  = R[4]; out[6]  = R[5]; out[7]  = t[1];
    out[8]  = R[6]; out[9]  = R[7]; out[10] = R[8]; out[11] = t[2];
    out[12] = 0.f;  out[13] = 0.f;  out[14] = 0.f;  out[15] = 1.f;
  }
}

// ----------------------------------------------------------------------------
extern "C" void kernel_launch(void* const* d_in, const int* in_sizes, int n_in,
                              void* d_out, int out_size, void* d_ws, size_t ws_size,
                              hipStream_t stream) {
  (void)in_sizes; (void)n_in; (void)out_size; (void)ws_size;
  const float* A = (const float*)d_in[0];
  const float* B = (const float*)d_in[1];
  float* ws  = (float*)d_ws;   // needs ~235 KB
  float* out = (float*)d_out;  // 16 floats (4x4 T, row-major)

  icp_init<<<BLK_A, 256, 0, stream>>>(A, B, ws);
  for (int it = 0; it < ITERS; ++it) {
    icp_nn<<<BLK_A, 256, 0, stream>>>(B, ws);
    icp_fit<<<1, 256, 0, stream>>>(ws);
  }
  icp_final<<<1, 256, 0, stream>>>(A, ws, out);
}